// LSTM_2714419331402
// MI455X (gfx1250) — compile-verified
//
#include <hip/hip_runtime.h>

typedef float v2f __attribute__((ext_vector_type(2)));
typedef float v8f __attribute__((ext_vector_type(8)));

#define I_DIM 256
#define H_DIM 512
#define O_DIM 10
#define B_DIM 64
#define T_DIM 2048
#define NWG   128                 // workgroups; each owns 4 h-columns
#define NC    4                   // h-columns per WG
#define GC    16                  // gate columns per WG (4 gates * NC)
#define KTOT  (H_DIM + I_DIM)     // 768 (fused [h | x_t] contraction)
#define KSTEPS (KTOT / 4)         // 192 WMMA k-steps (K=4 per fp32 WMMA)
#define KH    (H_DIM / 4)         // 128 k-steps sourced from h
#define THREADS 128               // 4 wave32 -> 4 M-tiles of 16 rows

__device__ __forceinline__ float fsigmoid(float x) {
    return 1.0f / (1.0f + __expf(-x));
}
__device__ __forceinline__ float ftanh(float x) {
    return 1.0f - 2.0f / (__expf(2.0f * x) + 1.0f);
}

__global__ __launch_bounds__(THREADS) void lstm_persistent(
    const float* __restrict__ x,      // [B, T, I]
    const float* __restrict__ W_ih,   // [4H, I]
    const float* __restrict__ W_hh,   // [4H, H]
    const float* __restrict__ b_ih,   // [4H]
    const float* __restrict__ b_hh,   // [4H]
    const float* __restrict__ W_fc,   // [O, H]
    const float* __restrict__ b_fc,   // [O]
    float* __restrict__ out,          // [B, O]
    float* __restrict__ ws)
{
    // Packed B operand: [s][ksel][n][pair] so each lane's fragment is one b64.
    __shared__ float Wp[KSTEPS * 2 * 16 * 2];   // 48 KB
    __shared__ float gbuf[B_DIM * GC];          // 4 KB gate slice [64 x 16]
    __shared__ float cbuf[B_DIM * NC];          // 1 KB cell state slice
    __shared__ float bias_s[GC];

    const int tid  = threadIdx.x;
    const int wg   = blockIdx.x;
    const int wave = tid >> 5;
    const int lane = tid & 31;
    const int nlo  = lane & 15;
    const int g2   = lane >> 4;     // half-wave selects K pair {2g2, 2g2+1}

    unsigned* barcnt = (unsigned*)ws;
    unsigned* bargen = barcnt + 1;
    float* hbuf = ws + 64;          // two [64 x 512] h buffers (double-buffered)

    // ---- one-time: pack this WG's weight slice into LDS in fragment order ----
    for (int idx = tid; idx < KSTEPS * 64; idx += THREADS) {
        int p  = idx & 1;            // element within lane pair
        int n  = (idx >> 1) & 15;    // column lane
        int gg = (idx >> 5) & 1;     // ksel group
        int s  = idx >> 6;           // k-step
        int k  = s * 4 + gg * 2 + p;
        int grow = (n >> 2) * H_DIM + wg * NC + (n & 3);  // gate row (i,f,g,o blocks)
        Wp[idx] = (k < H_DIM) ? W_hh[grow * H_DIM + k]
                              : W_ih[grow * I_DIM + (k - H_DIM)];
    }
    if (tid < GC) {
        int grow = (tid >> 2) * H_DIM + wg * NC + (tid & 3);
        bias_s[tid] = b_ih[grow] + b_hh[grow];
    }
    for (int idx = tid; idx < B_DIM * NC; idx += THREADS) cbuf[idx] = 0.0f;
    __syncthreads();

    const v2f* Wp2 = (const v2f*)Wp;
    const int m    = wave * 16 + nlo;   // A-matrix row this lane supplies
    const int kb0  = 2 * g2;

    for (int t = 0; t < T_DIM; ++t) {
        const float* hprev = hbuf + (size_t)(t & 1) * (B_DIM * H_DIM);
        float*       hnext = hbuf + (size_t)((t + 1) & 1) * (B_DIM * H_DIM);
        const float* hrow  = hprev + m * H_DIM;
        const float* xrow  = x + ((size_t)m * T_DIM + t) * I_DIM;

        // Pull x rows for t+2 toward the caches (covers HBM latency; x is the
        // only operand not produced on-device each step).
        __builtin_prefetch(xrow + 2 * I_DIM, 0, 1);

        v8f acc = {};
        #pragma unroll 8
        for (int s = 0; s < KH; ++s) {            // K from h
            v2f a = *(const v2f*)(hrow + 4 * s + kb0);
            v2f b = Wp2[s * 32 + g2 * 16 + nlo];
            acc = __builtin_amdgcn_wmma_f32_16x16x4_f32(
                false, a, false, b, (short)0, acc, false, false);
        }
        #pragma unroll 8
        for (int s = KH; s < KSTEPS; ++s) {       // K from x_t (fused input proj)
            v2f a = *(const v2f*)(xrow + 4 * (s - KH) + kb0);
            v2f b = Wp2[s * 32 + g2 * 16 + nlo];
            acc = __builtin_amdgcn_wmma_f32_16x16x4_f32(
                false, a, false, b, (short)0, acc, false, false);
        }

        // Spill gate tile to LDS (C/D layout: vgpr r -> M=r (+8 for lanes 16-31))
        #pragma unroll
        for (int r = 0; r < 8; ++r) {
            int row = wave * 16 + r + 8 * g2;
            gbuf[row * GC + nlo] = acc[r];
        }
        __syncthreads();

        // Elementwise cell update: this WG's 64x4 slice of c and h.
        #pragma unroll
        for (int e = tid; e < B_DIM * NC; e += THREADS) {
            int b  = e >> 2, cc = e & 3;
            float iv = fsigmoid(gbuf[b * GC +      cc] + bias_s[     cc]);
            float fv = fsigmoid(gbuf[b * GC +  4 + cc] + bias_s[ 4 + cc]);
            float gv = ftanh   (gbuf[b * GC +  8 + cc] + bias_s[ 8 + cc]);
            float ov = fsigmoid(gbuf[b * GC + 12 + cc] + bias_s[12 + cc]);
            float cn = fv * cbuf[e] + iv * gv;
            cbuf[e] = cn;
            hnext[b * H_DIM + wg * NC + cc] = ov * ftanh(cn);
        }

        // ---- device-wide barrier (sense via generation counter) ----
        __syncthreads();
        if (tid == 0) {
            __builtin_amdgcn_fence(__ATOMIC_RELEASE, "agent");
            unsigned g = __hip_atomic_load(bargen, __ATOMIC_RELAXED,
                                           __HIP_MEMORY_SCOPE_AGENT);
            unsigned arrived = __hip_atomic_fetch_add(
                barcnt, 1u, __ATOMIC_ACQ_REL, __HIP_MEMORY_SCOPE_AGENT) + 1u;
            if (arrived == (unsigned)NWG) {
                __hip_atomic_store(barcnt, 0u, __ATOMIC_RELAXED,
                                   __HIP_MEMORY_SCOPE_AGENT);
                __hip_atomic_fetch_add(bargen, 1u, __ATOMIC_RELEASE,
                                       __HIP_MEMORY_SCOPE_AGENT);
            } else {
                while (__hip_atomic_load(bargen, __ATOMIC_RELAXED,
                                         __HIP_MEMORY_SCOPE_AGENT) == g)
                    __builtin_amdgcn_s_sleep(1);
            }
            __builtin_amdgcn_fence(__ATOMIC_ACQUIRE, "agent");
        }
        __syncthreads();
    }

    // ---- final FC: h_last @ W_fcT + b_fc (tiny: 64x10), WG 0 only ----
    if (wg == 0) {
        const float* hfin = hbuf + (size_t)(T_DIM & 1) * (B_DIM * H_DIM);
        for (int e = tid; e < B_DIM * O_DIM; e += THREADS) {
            int b = e / O_DIM, o = e % O_DIM;
            float s = b_fc[o];
            for (int k = 0; k < H_DIM; ++k)
                s += hfin[b * H_DIM + k] * W_fc[o * H_DIM + k];
            out[e] = s;
        }
    }
}

extern "C" void kernel_launch(void* const* d_in, const int* in_sizes, int n_in,
                              void* d_out, int out_size, void* d_ws, size_t ws_size,
                              hipStream_t stream) {
    const float* x    = (const float*)d_in[0];
    const float* W_ih = (const float*)d_in[1];
    const float* W_hh = (const float*)d_in[2];
    const float* b_ih = (const float*)d_in[3];
    const float* b_hh = (const float*)d_in[4];
    const float* W_fc = (const float*)d_in[5];
    const float* b_fc = (const float*)d_in[6];
    float* out = (float*)d_out;
    float* ws  = (float*)d_ws;

    // ws layout: [0,256) barrier state | [256, 256+128KB) h buffer 0 | h buffer 1
    // Zero barrier + initial hidden state each launch (capture-legal memset).
    hipMemsetAsync(d_ws, 0, 256 + (size_t)B_DIM * H_DIM * sizeof(float), stream);

    lstm_persistent<<<NWG, THREADS, 0, stream>>>(
        x, W_ih, W_hh, b_ih, b_hh, W_fc, b_fc, out, ws);
}